// ComposeLoss_77730318123291
// MI455X (gfx1250) — compile-verified
//
#include <hip/hip_runtime.h>
#include <hip/hip_bf16.h>
#include <math.h>

// ---------------------------------------------------------------------------
// ComposeLoss for MI455X (gfx1250, wave32).
// Focal part: memory-bound stream of 32MB (heat+targets) -> ~1.5us at 23.3TB/s.
// Reductions use V_WMMA_F32_16X16X4_F32 with an all-ones B matrix as a
// 64-way wave-level summation unit (D[m][n] = sum_k A[m][k]); accumulator
// chaining (D->C) across iterations. All reductions are fixed-order
// (deterministic): per-thread sequential, shfl_xor tree, WMMA, no atomics.
// ---------------------------------------------------------------------------

typedef float v2f __attribute__((ext_vector_type(2)));
typedef float v4f __attribute__((ext_vector_type(4)));
typedef float v8f __attribute__((ext_vector_type(8)));
typedef int   v2i __attribute__((ext_vector_type(2)));

#define B_   16
#define H_   512
#define W_   512
#define HW_  (H_ * W_)     // 262144
#define N_   32
#define P_   128
#define NB_  64            // partial-blocks per batch image
#define TPB_ 256           // 8 wave32 per block
#define CH_  (HW_ / NB_)   // 4096 elements per block

__device__ __forceinline__ float wave_sum32(float v) {
#pragma unroll
  for (int off = 16; off > 0; off >>= 1) v += __shfl_xor(v, off, 32);
  return v;
}

// Sum 64 f32 values (2 per lane across wave32) on the matrix unit.
// D = A(16x4) x ones(4x16): every slot of A contributes exactly once to the
// row-sums; summing the 8 accumulator VGPRs per lane and folding lane n with
// lane n+16 yields the grand total in every lane.
__device__ __forceinline__ float wmma_sum64(v2f a) {
  v2f ones; ones[0] = 1.0f; ones[1] = 1.0f;
  v8f c = {0.0f, 0.0f, 0.0f, 0.0f, 0.0f, 0.0f, 0.0f, 0.0f};
  v8f d = __builtin_amdgcn_wmma_f32_16x16x4_f32(
      false, a, false, ones, (short)0, c, false, false);
  float s = d[0] + d[1] + d[2] + d[3] + d[4] + d[5] + d[6] + d[7];
  s += __shfl_xor(s, 16, 32);
  return s;
}

// ---------------------------------------------------------------------------
// Kernel 1: focal-loss partials. grid = B_*NB_ blocks x 256 threads.
// Each block handles 4096 contiguous pixels of one image via b128 NT loads.
// Writes ws[b*NB+j] = pos_sum, ws[1024+..] = neg_sum, ws[2048+..] = num_pos.
// ---------------------------------------------------------------------------
__global__ void __launch_bounds__(TPB_)
focal_partial_kernel(const float* __restrict__ heat,
                     const float* __restrict__ targ,
                     float* __restrict__ ws) {
  const int b = blockIdx.x / NB_;
  const int j = blockIdx.x % NB_;
  const long base = (long)b * HW_ + (long)j * CH_;
  const v4f* h4 = (const v4f*)(heat + base);
  const v4f* t4 = (const v4f*)(targ + base);

  float sp = 0.0f, sn = 0.0f, np = 0.0f;
  for (int i = threadIdx.x; i < CH_ / 4; i += TPB_) {
    v4f h = __builtin_nontemporal_load(h4 + i);   // single pass: NT hint
    v4f t = __builtin_nontemporal_load(t4 + i);
#pragma unroll
    for (int k = 0; k < 4; ++k) {
      float hh = h[k], tt = t[k];
      float pos = (tt == 1.0f) ? 1.0f : 0.0f;
      float sg = 1.0f / (1.0f + expf(-hh));
      float pred = fminf(fmaxf(sg, 1e-4f), 1.0f - 1e-4f);
      float pt = (pos != 0.0f) ? pred : (1.0f - pred);
      // (1-pt)^0.25 == sqrt(sqrt(1-pt)); 1-pt >= EPS > 0
      float loss = -sqrtf(sqrtf(1.0f - pt)) * logf(pt);
      sp += 0.9f * loss * pos;                       // alpha * loss * pos
      float omt = 1.0f - tt;
      sn += omt * omt * (0.1f * loss) * (1.0f - pos); // (1-t)^2*(1-a)*loss*(1-pos)
      np += pos;
    }
  }

  sp = wave_sum32(sp);
  sn = wave_sum32(sn);
  np = wave_sum32(np);

  __shared__ float red[8][3];
  const int wv = threadIdx.x >> 5;
  if ((threadIdx.x & 31) == 0) {
    red[wv][0] = sp; red[wv][1] = sn; red[wv][2] = np;
  }
  __syncthreads();
  if (threadIdx.x == 0) {
    float a = 0.0f, c = 0.0f, d = 0.0f;
#pragma unroll
    for (int w = 0; w < 8; ++w) { a += red[w][0]; c += red[w][1]; d += red[w][2]; }
    const int idx = b * NB_ + j;
    ws[idx]                = a;
    ws[B_ * NB_ + idx]     = c;
    ws[2 * B_ * NB_ + idx] = d;
  }
}

// ---------------------------------------------------------------------------
// Kernel 2: AE loss per polygon. grid = B_*N_ = 512 blocks x 32 threads.
// Each lane handles 4 points (coalesced int2 loads of (y,x)).
// ---------------------------------------------------------------------------
__global__ void __launch_bounds__(32)
ae_poly_kernel(const float* __restrict__ ae,
               const int* __restrict__ poly,
               float* __restrict__ out) {
  const int g = blockIdx.x;        // b*N + n
  const int b = g / N_;
  const int lane = threadIdx.x;
  const v2i* pp = (const v2i*)(poly + (long)g * P_ * 2);
  const float* ae0 = ae + (long)b * 2 * HW_;
  const float* ae1 = ae0 + HW_;

  int ys[4], xs[4];
  float a0[4], a1[4];
  float sy = 0.0f, sx = 0.0f;
#pragma unroll
  for (int k = 0; k < 4; ++k) {
    v2i q = pp[lane + 32 * k];     // (y, x)
    ys[k] = q[0]; xs[k] = q[1];
    const int off = ys[k] * W_ + xs[k];
    a0[k] = ae0[off];
    a1[k] = ae1[off];
    sy += (float)ys[k];
    sx += (float)xs[k];
  }
  sy = wave_sum32(sy);
  sx = wave_sum32(sx);
  const float cy = floorf(sy * (1.0f / (float)P_));
  const float cx = floorf(sx * (1.0f / (float)P_));

  float s = 0.0f;
#pragma unroll
  for (int k = 0; k < 4; ++k) {
    float d0 = a0[k] + (float)ys[k] - cy;
    float d1 = a1[k] + (float)xs[k] - cx;
    s += sqrtf(d0 * d0 + d1 * d1);
  }
  s = wave_sum32(s);
  if (lane == 0) out[g] = s * (1.0f / (float)P_);
}

// ---------------------------------------------------------------------------
// Kernel 3: per-batch reduction of the 64 focal partials via WMMA.
// grid = 16 blocks x 32 threads (one wave each; EXEC all-ones for WMMA).
// ---------------------------------------------------------------------------
__global__ void __launch_bounds__(32)
batch_reduce_kernel(const float* __restrict__ ws, float* __restrict__ kpb) {
  const int b = blockIdx.x;
  const int lane = threadIdx.x;
  const v2f* pp = (const v2f*)(ws + b * NB_);
  const v2f* pn = (const v2f*)(ws + B_ * NB_ + b * NB_);
  const v2f* pz = (const v2f*)(ws + 2 * B_ * NB_ + b * NB_);
  v2f ap = pp[lane];
  v2f an = pn[lane];
  v2f az = pz[lane];
  float sp = wmma_sum64(ap);
  float sn = wmma_sum64(an);
  float sz = wmma_sum64(az);
  if (lane == 0) {
    // norm = max(0.9*100 + 0.1*num_pos, 1)
    float norm = fmaxf(90.0f + 0.1f * sz, 1.0f);
    kpb[b] = (sp + sn) / norm;
  }
}

// ---------------------------------------------------------------------------
// Kernel 4: final combine. 1 block x 32 threads.
// kp = mean over B of kpb; ae = 0.1 * mean over 512 per-polygon values,
// the latter summed with 8 chained accumulating WMMAs (D->C).
// ---------------------------------------------------------------------------
__global__ void __launch_bounds__(32)
final_combine_kernel(const float* __restrict__ kpb,
                     const float* __restrict__ polyv,
                     float* __restrict__ out) {
  const int lane = threadIdx.x;
  // each kpb[i] read by lanes i and i+16 -> wave sum counts each twice
  float k = kpb[lane & 15];
  k = wave_sum32(k) * (1.0f / 32.0f);   // == mean over 16

  v2f ones; ones[0] = 1.0f; ones[1] = 1.0f;
  v8f acc = {0.0f, 0.0f, 0.0f, 0.0f, 0.0f, 0.0f, 0.0f, 0.0f};
  const v2f* pv = (const v2f*)polyv;
#pragma unroll
  for (int it = 0; it < (B_ * N_) / 64; ++it) {   // 512/64 = 8 iterations
    v2f a = pv[it * 32 + lane];
    acc = __builtin_amdgcn_wmma_f32_16x16x4_f32(
        false, a, false, ones, (short)0, acc, false, false);
  }
  float s = acc[0] + acc[1] + acc[2] + acc[3] + acc[4] + acc[5] + acc[6] + acc[7];
  s += __shfl_xor(s, 16, 32);

  if (lane == 0) out[0] = k + 0.1f * s * (1.0f / (float)(B_ * N_));
}

// ---------------------------------------------------------------------------
extern "C" void kernel_launch(void* const* d_in, const int* in_sizes, int n_in,
                              void* d_out, int out_size, void* d_ws, size_t ws_size,
                              hipStream_t stream) {
  const float* kp_heat    = (const float*)d_in[0];  // [16,1,512,512] f32
  const float* ae_map     = (const float*)d_in[1];  // [16,2,512,512] f32
  const float* kp_targets = (const float*)d_in[2];  // [16,1,512,512] f32
  const int*   polygons   = (const int*)d_in[3];    // [16,32,128,2] i32
  float* out = (float*)d_out;                       // scalar

  // workspace layout (floats): [0,3072) focal partials (pos/neg/np x 1024),
  // [3072,3088) per-batch kp, [3088,3600) per-polygon AE values.
  float* ws       = (float*)d_ws;
  float* partials = ws;
  float* kpb      = ws + 3 * B_ * NB_;
  float* polyv    = kpb + B_;

  focal_partial_kernel<<<dim3(B_ * NB_), dim3(TPB_), 0, stream>>>(
      kp_heat, kp_targets, partials);
  ae_poly_kernel<<<dim3(B_ * N_), dim3(32), 0, stream>>>(
      ae_map, polygons, polyv);
  batch_reduce_kernel<<<dim3(B_), dim3(32), 0, stream>>>(partials, kpb);
  final_combine_kernel<<<dim3(1), dim3(32), 0, stream>>>(kpb, polyv, out);
}